// GCN_81406810128958
// MI455X (gfx1250) — compile-verified
//
#include <hip/hip_runtime.h>
#include <hip/hip_bf16.h>

typedef __attribute__((ext_vector_type(2))) float v2f;
typedef __attribute__((ext_vector_type(4))) float f4;
typedef __attribute__((ext_vector_type(8))) float v8f;

// ---------------------------------------------------------------------------
// Kernel 1: single-wave GCN on the tensor core (V_WMMA_F32_16X16X4_F32).
// All matrices are zero-padded to 16x16 tiles held in LDS; intermediates are
// written back to LDS between WMMAs so operand layouts are always rebuilt
// from the documented lane mappings:
//   A (16x4 f32):  lane<16 -> row=lane, {v0,v1}={K,K+1}; lane>=16 -> {K+2,K+3}
//   B (4x16 f32):  lane%16 = col,      {v0,v1}={K,K+1} (low half) / {K+2,K+3}
//   C/D (16x16):   vgpr r, lane -> row r + 8*(lane/16), col lane%16
// ---------------------------------------------------------------------------

__device__ __forceinline__ v8f wmma_tile(const float* Abase, int lda, int ktot,
                                         const float* Bbase, int ldb) {
    const int lane = threadIdx.x & 31;
    const int half = lane >> 4;
    const int l    = lane & 15;
    v8f c = {};
    for (int k0 = 0; k0 < ktot; k0 += 4) {
        const int ka = k0 + 2 * half;
        v2f a, b;
        a.x = Abase[l * lda + ka];
        a.y = Abase[l * lda + ka + 1];
        b.x = Bbase[ka * ldb + l];
        b.y = Bbase[(ka + 1) * ldb + l];
        c = __builtin_amdgcn_wmma_f32_16x16x4_f32(
                /*neg_a=*/false, a, /*neg_b=*/false, b,
                /*c_mod=*/(short)0, c, /*reuse_a=*/false, /*reuse_b=*/false);
    }
    return c;
}

__device__ __forceinline__ void store_tile(float* Cbase, int ldc, v8f c, bool relu) {
    const int lane = threadIdx.x & 31;
    const int half = lane >> 4;
    const int l    = lane & 15;
#pragma unroll
    for (int r = 0; r < 8; ++r) {
        float v = c[r];
        if (relu) v = v > 0.0f ? v : 0.0f;
        Cbase[(r + 8 * half) * ldc + l] = v;
    }
}

// Load W^T (W is [nout, nin] row-major; Wt[k][n] = W[n*nin + k]) into a
// zero-padded 32x32 LDS tile.
__device__ __forceinline__ void load_wt(float* sW, const float* W,
                                        int nin, int nout, int lane) {
    for (int i = lane; i < 32 * 32; i += 32) {
        const int k = i >> 5;
        const int n = i & 31;
        sW[i] = (k < nin && n < nout) ? W[n * nin + k] : 0.0f;
    }
}

__global__ __launch_bounds__(32) void gcn_node_kernel(
        const float* __restrict__ emb, const float* __restrict__ A,
        const float* __restrict__ W0,  const float* __restrict__ W1,
        const float* __restrict__ W2,  const float* __restrict__ Wf,
        const float* __restrict__ bf,  float* __restrict__ lut /* [6][16] */) {
    __shared__ float sA[16 * 16];   // A padded (5x5 -> 16x16)
    __shared__ float sX[16 * 32];   // activation ping
    __shared__ float sY[16 * 32];   // activation pong
    __shared__ float sW[32 * 32];   // current transposed weight tile

    const int lane = threadIdx.x & 31;

    // ---- stage A (5x5) and E (5x16) zero-padded ----
    for (int i = lane; i < 256; i += 32) {
        const int r = i >> 4, c = i & 15;
        sA[i] = (r < 5 && c < 5) ? A[r * 5 + c] : 0.0f;
    }
    for (int i = lane; i < 512; i += 32) {
        const int r = i >> 5, c = i & 31;
        sX[i] = (r < 5 && c < 16) ? emb[r * 16 + c] : 0.0f;
    }
    __syncthreads();

    // x1 = A @ E            [16x16], K=5 padded to 8
    v8f c0 = wmma_tile(sA, 16, 8, sX, 32);
    store_tile(sY, 32, c0, false);
    __syncthreads();

    load_wt(sW, W0, 16, 32, lane);          // W0^T : 16x32
    __syncthreads();

    // h0 = relu(x1 @ W0^T)  [16x32], K=16, two N tiles
    v8f ha = wmma_tile(sY, 32, 16, sW, 32);
    v8f hb = wmma_tile(sY, 32, 16, sW + 16, 32);
    store_tile(sX, 32, ha, true);
    store_tile(sX + 16, 32, hb, true);
    __syncthreads();

    // t = A @ h0            [16x32], K=8
    v8f ta = wmma_tile(sA, 16, 8, sX, 32);
    v8f tb = wmma_tile(sA, 16, 8, sX + 16, 32);
    store_tile(sY, 32, ta, false);
    store_tile(sY + 16, 32, tb, false);
    __syncthreads();

    load_wt(sW, W1, 32, 32, lane);          // W1^T : 32x32
    __syncthreads();

    // h1 = relu(t @ W1^T)   [16x32], K=32
    ha = wmma_tile(sY, 32, 32, sW, 32);
    hb = wmma_tile(sY, 32, 32, sW + 16, 32);
    store_tile(sX, 32, ha, true);
    store_tile(sX + 16, 32, hb, true);
    __syncthreads();

    // t2 = A @ h1           [16x32], K=8
    ta = wmma_tile(sA, 16, 8, sX, 32);
    tb = wmma_tile(sA, 16, 8, sX + 16, 32);
    store_tile(sY, 32, ta, false);
    store_tile(sY + 16, 32, tb, false);
    __syncthreads();

    load_wt(sW, W2, 32, 32, lane);          // W2^T : 32x32
    __syncthreads();

    // h2 = relu(t2 @ W2^T)  [16x32], K=32
    ha = wmma_tile(sY, 32, 32, sW, 32);
    hb = wmma_tile(sY, 32, 32, sW + 16, 32);
    store_tile(sX, 32, ha, true);
    store_tile(sX + 16, 32, hb, true);
    __syncthreads();

    load_wt(sW, Wf, 32, 16, lane);          // Wf^T : 32x16
    __syncthreads();

    // node_emb = h2 @ Wf^T + bf   [16x16], K=32 (rows 0..4 meaningful)
    v8f fin = wmma_tile(sX, 32, 32, sW, 32);

    const int half = lane >> 4;
    const int l    = lane & 15;
#pragma unroll
    for (int r = 0; r < 8; ++r) {
        const int m = r + 8 * half;
        if (m < 5) lut[(1 + m) * 16 + l] = fin[r] + bf[l];  // rows 1..5 = node_emb
    }
    if (lane < 16) lut[lane] = emb[lane];                   // row 0 = emb_table[0]
}

// ---------------------------------------------------------------------------
// Kernel 2: memory-bound scatter.  out[cell] = lut[game_state[cell]] (16 f32).
// 272 MB of mandatory traffic -> pure HBM-store-bandwidth problem; each thread
// does one int32 load, 4 ds_load_b128 (broadcast-friendly), and 4 non-temporal
// global_store_b128.
// ---------------------------------------------------------------------------
__global__ __launch_bounds__(256) void scatter_kernel(
        const int* __restrict__ gs, const float* __restrict__ lut,
        float* __restrict__ out, int ncells) {
    __shared__ float slut[6 * 16];
    if (threadIdx.x < 96) slut[threadIdx.x] = lut[threadIdx.x];
    __syncthreads();

    const int idx = blockIdx.x * blockDim.x + threadIdx.x;
    if (idx < ncells) {
        int g = gs[idx];
        g = (g < 0) ? 0 : (g > 5 ? 5 : g);
        const f4* row = (const f4*)(slut + g * 16);
        f4* o = (f4*)(out + (size_t)idx * 16);
        __builtin_nontemporal_store(row[0], o + 0);
        __builtin_nontemporal_store(row[1], o + 1);
        __builtin_nontemporal_store(row[2], o + 2);
        __builtin_nontemporal_store(row[3], o + 3);
    }
}

extern "C" void kernel_launch(void* const* d_in, const int* in_sizes, int n_in,
                              void* d_out, int out_size, void* d_ws, size_t ws_size,
                              hipStream_t stream) {
    const int*   gs  = (const int*)d_in[0];
    const float* emb = (const float*)d_in[1];
    const float* A   = (const float*)d_in[2];
    const float* W0  = (const float*)d_in[3];
    const float* W1  = (const float*)d_in[4];
    const float* W2  = (const float*)d_in[5];
    const float* Wf  = (const float*)d_in[6];
    const float* bf  = (const float*)d_in[7];
    float* out = (float*)d_out;
    float* lut = (float*)d_ws;          // 6*16 floats = 384 B

    const int ncells = in_sizes[0];     // 256*128*128

    gcn_node_kernel<<<1, 32, 0, stream>>>(emb, A, W0, W1, W2, Wf, bf, lut);
    scatter_kernel<<<(ncells + 255) / 256, 256, 0, stream>>>(gs, lut, out, ncells);
}